// SSLinear_17051020165438
// MI455X (gfx1250) — compile-verified
//
#include <hip/hip_runtime.h>
#include <cstdint>

// SSLinear on MI455X (gfx1250): 2:4-sparse weight x row-sparse activation GEMM
// using V_SWMMAC_F32_16X16X64_F16 (wave32).

typedef __attribute__((ext_vector_type(16))) _Float16 v16h;
typedef __attribute__((ext_vector_type(32))) _Float16 v32h;
typedef __attribute__((ext_vector_type(8)))  float    v8f;

#define O_DIM   4096   // compressed out rows
#define I_DIM   2048   // compressed k
#define M_FULL  8192   // dense out rows
#define K_FULL  4096   // dense k
#define NTOK    4096   // tokens (GEMM N)
#define K_ACT   2048   // active activation rows
#define TSTEPS  64     // K_FULL / 64 dense-K per swmmac
#define ATILES  256    // O_DIM / 16
#define BTILES  256    // NTOK  / 16

// ---------------- workspace layout (bytes) ----------------
#define WS_INV_OFF   0u                       // int  [K_FULL]
#define WS_APACK_OFF 16384u                   // f16  [ATILES][TSTEPS][32][16]
#define WS_META_OFF  (16384u + 16777216u)     // u32  [ATILES][TSTEPS][32]
#define WS_BPACK_OFF (16384u + 16777216u + 2097152u) // f16 [TSTEPS][BTILES][32][32]

__global__ void zero_out_kernel(float4* __restrict__ out, long n4) {
  long p = blockIdx.x * (long)blockDim.x + threadIdx.x;
  if (p < n4) out[p] = make_float4(0.f, 0.f, 0.f, 0.f);
}

// inv[kd] = j such that idx[j]==kd, else -1  (idx sorted unique)
__global__ void build_inv_kernel(const int* __restrict__ idx, int* __restrict__ inv) {
  int kd = blockIdx.x * blockDim.x + threadIdx.x;
  if (kd >= K_FULL) return;
  int lo = 0, hi = K_ACT - 1, res = -1;
  while (lo <= hi) {
    int mid = (lo + hi) >> 1;
    int v = idx[mid];
    if (v == kd) { res = mid; break; }
    else if (v < kd) lo = mid + 1;
    else hi = mid - 1;
  }
  inv[kd] = res;
}

// Pack compressed weight into SWMMAC A layout (16x32 f16 stored tile):
// lane<16: row=tile*16+lane, Kc = e<8 ? e : e+8
// lane>=16: same rows,       Kc = (e<8 ? e : e+8) + 8
__global__ void pack_a_kernel(const float* __restrict__ w, _Float16* __restrict__ ap) {
  long p = blockIdx.x * (long)blockDim.x + threadIdx.x;
  if (p >= (long)ATILES * TSTEPS * 32 * 16) return;
  int e    = (int)(p & 15);
  int lane = (int)((p >> 4) & 31);
  int t    = (int)((p >> 9) & 63);
  int tile = (int)(p >> 15);
  int row  = tile * 16 + (lane & 15);
  int kc   = ((e < 8) ? e : e + 8) + ((lane >= 16) ? 8 : 0);
  int i    = t * 32 + kc;                       // compressed-k index
  ap[p] = (_Float16)w[(long)row * I_DIM + i];
}

// Pack metadata into SWMMAC index VGPR: per lane a 32-bit word of 8 nibbles,
// nibble j = idx0 | (idx1<<2) for dense-col group (t*16 + (lane>=16?8:0) + j).
__global__ void pack_meta_kernel(const int* __restrict__ md, unsigned* __restrict__ mp) {
  int p = blockIdx.x * blockDim.x + threadIdx.x;
  if (p >= ATILES * TSTEPS * 32) return;
  int lane = p & 31;
  int t    = (p >> 5) & 63;
  int tile = p >> 11;
  int row  = tile * 16 + (lane & 15);
  int gbase = t * 16 + ((lane >= 16) ? 8 : 0);
  const int* mrow = md + (long)row * I_DIM;
  unsigned word = 0u;
  #pragma unroll
  for (int j = 0; j < 8; ++j) {
    int g = gbase + j;
    unsigned n0 = (unsigned)mrow[2 * g]     & 3u;
    unsigned n1 = (unsigned)mrow[2 * g + 1] & 3u;
    word |= (n0 | (n1 << 2)) << (4 * j);
  }
  mp[p] = word;
}

// Pack dense-equivalent B (64x16 f16 tile) gathered from input via inv:
// lane -> N = ntile*16 + lane%16 ; element e -> K = (e/16)*32 + (lane>=16?16:0) + e%16
__global__ void pack_b_kernel(const float* __restrict__ inp, const int* __restrict__ inv,
                              _Float16* __restrict__ bp) {
  long p = blockIdx.x * (long)blockDim.x + threadIdx.x;
  if (p >= (long)TSTEPS * BTILES * 32 * 32) return;
  int e     = (int)(p & 31);
  int lane  = (int)((p >> 5) & 31);
  int ntile = (int)((p >> 10) & 255);
  int t     = (int)(p >> 18);
  int nn    = ntile * 16 + (lane & 15);
  int kd    = t * 64 + (e >> 4) * 32 + ((lane >= 16) ? 16 : 0) + (e & 15);
  int j     = inv[kd];
  float v   = (j >= 0) ? inp[(long)nn * K_ACT + j] : 0.0f;
  bp[p] = (_Float16)v;
}

// One wave computes a 64x64 output tile: 4x4 sub-tiles of 16x16, K loop of 64
// dense-K steps, each step = 16 v_swmmac_f32_16x16x64_f16 with A/index reuse.
__global__ void __launch_bounds__(128, 1)
sswmmac_gemm_kernel(const _Float16* __restrict__ ap, const unsigned* __restrict__ mp,
                    const _Float16* __restrict__ bp, const int* __restrict__ indices,
                    float* __restrict__ out) {
  const int lane   = threadIdx.x & 31;
  const int wave   = threadIdx.x >> 5;
  const int tileId = blockIdx.x * 4 + wave;      // 4096 tiles of 64x64
  const int ot     = tileId >> 6;                // o-block  (units of 64 rows)
  const int nt     = tileId & 63;                // n-block  (units of 64 cols)

  v8f acc[4][4];
  #pragma unroll
  for (int mo = 0; mo < 4; ++mo)
    #pragma unroll
    for (int no = 0; no < 4; ++no)
      acc[mo][no] = (v8f){0.f, 0.f, 0.f, 0.f, 0.f, 0.f, 0.f, 0.f};

  for (int t = 0; t < TSTEPS; ++t) {
    v16h a[4];
    int  mi[4];
    #pragma unroll
    for (int mo = 0; mo < 4; ++mo) {
      const long atile = (long)(ot * 4 + mo);
      const long abase = ((atile * TSTEPS + t) * 32 + lane);
      a[mo]  = *(const v16h*)(ap + abase * 16);
      mi[mo] = (int)mp[abase];
    }
    #pragma unroll
    for (int no = 0; no < 4; ++no) {
      const long bbase = (((long)t * BTILES + (nt * 4 + no)) * 32 + lane);
      v32h b = *(const v32h*)(bp + bbase * 32);
      #pragma unroll
      for (int mo = 0; mo < 4; ++mo) {
        // D = A(2:4 sparse, 16x64 expanded) * B(64x16) + C
        acc[mo][no] = __builtin_amdgcn_swmmac_f32_16x16x64_f16(
            /*neg_a=*/false, a[mo], /*neg_b=*/false, b,
            acc[mo][no], mi[mo], /*reuse_a=*/false, /*reuse_b=*/false);
      }
    }
  }

  // C/D layout: VGPR r, lanes 0-15 -> (M=r, N=lane); lanes 16-31 -> (M=r+8, N=lane-16)
  const int ncol0 = nt * 64 + (lane & 15);
  const int mhalf = (lane >> 4) * 8;
  #pragma unroll
  for (int mo = 0; mo < 4; ++mo) {
    #pragma unroll
    for (int r = 0; r < 8; ++r) {
      const int ocomp = ot * 64 + mo * 16 + mhalf + r;
      const int orow  = indices[ocomp];          // dense output row id
      #pragma unroll
      for (int no = 0; no < 4; ++no) {
        out[(long)orow * NTOK + ncol0 + no * 16] = acc[mo][no][r];
      }
    }
  }
}

extern "C" void kernel_launch(void* const* d_in, const int* in_sizes, int n_in,
                              void* d_out, int out_size, void* d_ws, size_t ws_size,
                              hipStream_t stream) {
  const float* inp      = (const float*)d_in[0];   // [NTOK, K_ACT]
  const int*   idx      = (const int*)  d_in[1];   // [K_ACT]
  const float* weight   = (const float*)d_in[2];   // [O_DIM, I_DIM]
  const int*   indices  = (const int*)  d_in[3];   // [O_DIM]
  const int*   metadata = (const int*)  d_in[4];   // [O_DIM, I_DIM]
  float*       out      = (float*)d_out;           // [M_FULL, NTOK]

  char* ws = (char*)d_ws;
  int*      inv = (int*)     (ws + WS_INV_OFF);
  _Float16* apk = (_Float16*)(ws + WS_APACK_OFF);
  unsigned* mpk = (unsigned*)(ws + WS_META_OFF);
  _Float16* bpk = (_Float16*)(ws + WS_BPACK_OFF);

  // 1) zero the full output (rows not in `indices` must be zero)
  {
    long n4 = (long)M_FULL * NTOK / 4;
    zero_out_kernel<<<(int)((n4 + 255) / 256), 256, 0, stream>>>((float4*)out, n4);
  }
  // 2) inverse map of active activation rows
  build_inv_kernel<<<K_FULL / 256, 256, 0, stream>>>(idx, inv);
  // 3) pack A (f16, SWMMAC compressed-A layout)
  {
    long n = (long)ATILES * TSTEPS * 32 * 16;
    pack_a_kernel<<<(int)((n + 255) / 256), 256, 0, stream>>>(weight, apk);
  }
  // 4) pack 2:4 metadata into SWMMAC index words
  pack_meta_kernel<<<(ATILES * TSTEPS * 32) / 256, 256, 0, stream>>>(metadata, mpk);
  // 5) pack B (f16, SWMMAC B layout, gathered through inv)
  {
    long n = (long)TSTEPS * BTILES * 32 * 32;
    pack_b_kernel<<<(int)((n + 255) / 256), 256, 0, stream>>>(inp, inv, bpk);
  }
  // 6) sparse GEMM: 4096 waves, 4 waves per block
  sswmmac_gemm_kernel<<<1024, 128, 0, stream>>>(apk, mpk, bpk, indices, out);
}